// Euclidian_62302795596399
// MI455X (gfx1250) — compile-verified
//
#include <hip/hip_runtime.h>

// Problem shape (static per reference)
#define N_SAMPLES 500000
#define DIMS      256
#define NCLS      100
#define N_TILES   (N_SAMPLES / 16)          // 31250, exact

// Workspace layout (floats), all offsets 128-float aligned
#define SUMS_SZ     (NCLS * DIMS)           // 25600 class sums
#define CNT_OFF     SUMS_SZ                 // 100 counts (+pad)
#define TABLE_SZ    (SUMS_SZ + 128)         // 25728
#define CENTERS_OFF TABLE_SZ
#define SQC_OFF     (CENTERS_OFF + SUMS_SZ) // ||c||^2 per class
// total ws use: (SQC_OFF + 128) * 4 bytes  ~= 206 KB (tiny, safe)

typedef float v2f __attribute__((ext_vector_type(2)));
typedef float v8f __attribute__((ext_vector_type(8)));

__global__ void zero_ws(float* __restrict__ ws) {
    int i = blockIdx.x * blockDim.x + threadIdx.x;
    if (i < TABLE_SZ) ws[i] = 0.0f;
}

// Pass 1: segment-sum into a 100x256 f32 LDS table (100 KB of the WGP's 320 KB).
// Lane L owns dims {L, L+32, ..., L+224}: global loads are fully coalesced
// (128B/instr) and ds_add_f32 hits 32 distinct banks (stride 32 mod 64 banks).
__global__ __launch_bounds__(256) void accum_sums(const float* __restrict__ feat,
                                                  const int* __restrict__ label,
                                                  float* __restrict__ ws) {
    extern __shared__ float lds[];  // TABLE_SZ floats
    const int tid = threadIdx.x;
    for (int i = tid; i < TABLE_SZ; i += 256) lds[i] = 0.0f;
    __syncthreads();

    const int lane   = tid & 31;
    const int wave   = tid >> 5;
    const int gw     = blockIdx.x * 8 + wave;   // one sample row per wave/iter
    const int stride = gridDim.x * 8;

    for (int s = gw; s < N_SAMPLES; s += stride) {
        const int l = label[s];
        const float* row = feat + (size_t)s * DIMS;
        float x[8];
#pragma unroll
        for (int j = 0; j < 8; ++j) x[j] = row[j * 32 + lane];  // coalesced
        float* dst = lds + l * DIMS + lane;
#pragma unroll
        for (int j = 0; j < 8; ++j) atomicAdd(dst + j * 32, x[j]);  // ds_add_f32
        if (lane == 0) atomicAdd(&lds[CNT_OFF + l], 1.0f);
    }
    __syncthreads();

    // Flush block-partial table with native global fp32 atomics (skip zeros).
    for (int i = tid; i < TABLE_SZ; i += 256) {
        float v = lds[i];
        if (v != 0.0f) unsafeAtomicAdd(ws + i, v);
    }
}

// Pass 1b: centers = sum / count; sq_c = ||center||^2 (block-tree reduction).
__global__ __launch_bounds__(256) void finalize_centers(float* __restrict__ ws) {
    __shared__ float red[256];
    const int c = blockIdx.x;
    const int d = threadIdx.x;
    const float cnt = ws[CNT_OFF + c];
    const float inv = (cnt > 0.0f) ? (1.0f / cnt) : 0.0f;
    const float ctr = ws[c * DIMS + d] * inv;
    ws[CENTERS_OFF + c * DIMS + d] = ctr;
    red[d] = ctr * ctr;
    __syncthreads();
    for (int off = 128; off > 0; off >>= 1) {
        if (d < off) red[d] += red[d + off];
        __syncthreads();
    }
    if (d == 0) ws[SQC_OFF + c] = red[0];
}

// Pass 2: one wave per 16-sample tile. A = 16 feat rows, B = 16 gathered
// centers (columns). 64x V_WMMA_F32_16X16X4_F32 accumulate G = A*B; the
// diagonal G[m][m] = f_m . c_{l_m}. d2 = |f|^2 - 2g + |c|^2 (reference formula).
__global__ __launch_bounds__(256) void dist_wmma(const float* __restrict__ feat,
                                                 const int* __restrict__ label,
                                                 const float* __restrict__ ws,
                                                 float* __restrict__ out) {
    const int tid  = threadIdx.x;
    const int lane = tid & 31;
    const int tile = blockIdx.x * 8 + (tid >> 5);
    if (tile >= N_TILES) return;  // wave-uniform: EXEC stays all-ones for WMMA

    const float* centers = ws + CENTERS_OFF;
    const float* sqc     = ws + SQC_OFF;

    const int m    = lane & 15;     // sample row (lanes L and L+16 share a row)
    const int half = lane >> 4;     // 0: K={0,1};  1: K={2,3} of each chunk
    const int s    = tile * 16 + m;
    const int l    = label[s];

    const float* arow = feat + (size_t)s * DIMS;
    const float* brow = centers + l * DIMS;   // gathered center column

    v8f   acc = {};
    float sqf = 0.0f;

#pragma unroll 8
    for (int k0 = 0; k0 < DIMS; k0 += 4) {
        // A layout (16x4 f32): VGPR0 = K0|K2, VGPR1 = K1|K3 across lane halves
        v2f a = *(const v2f*)(arow + k0 + half * 2);
        v2f b = *(const v2f*)(brow + k0 + half * 2);
        sqf += a.x * a.x + a.y * a.y;
        acc = __builtin_amdgcn_wmma_f32_16x16x4_f32(
            /*neg_a=*/false, a, /*neg_b=*/false, b,
            /*c_mod=*/(short)0, acc, /*reuse_a=*/false, /*reuse_b=*/false);
    }

    // Lanes m and m+16 each summed half the row's squares.
    sqf += __shfl_xor(sqf, 16, 32);

    // Diagonal of the 16x16 f32 D tile: element (m,m) lives at
    // VGPR m%8, lane m (m<8) or lane m+16 (m>=8).
    float g = 0.0f;
#pragma unroll
    for (int mm = 0; mm < 16; ++mm) {
        float t = __shfl(acc[mm & 7], (mm < 8) ? mm : (mm + 16), 32);
        if (lane == mm) g = t;
    }

    if (lane < 16) {
        float d2 = sqf - 2.0f * g + sqc[l];
        out[s] = sqrtf(fmaxf(d2, 0.0f));
    }
}

extern "C" void kernel_launch(void* const* d_in, const int* in_sizes, int n_in,
                              void* d_out, int out_size, void* d_ws, size_t ws_size,
                              hipStream_t stream) {
    const float* feat  = (const float*)d_in[0];
    const int*   label = (const int*)d_in[1];
    float*       out   = (float*)d_out;
    float*       ws    = (float*)d_ws;

    zero_ws<<<(TABLE_SZ + 255) / 256, 256, 0, stream>>>(ws);

    // 256 persistent blocks, 8 waves each; dynamic LDS = 25728 floats (100.5 KB)
    accum_sums<<<256, 256, TABLE_SZ * sizeof(float), stream>>>(feat, label, ws);

    finalize_centers<<<NCLS, 256, 0, stream>>>(ws);

    dist_wmma<<<(N_TILES + 7) / 8, 256, 0, stream>>>(feat, label, ws, out);
}